// SSMBlock_3676492005424
// MI455X (gfx1250) — compile-verified
//
#include <hip/hip_runtime.h>
#include <math.h>

// ---- problem constants (match reference) ----
#define B_SZ    2
#define SEQ_L   2048
#define DMODEL  1024
#define DSTATE  16
#define DCONV   4
#define DINNER  2048            // EXPAND * DMODEL
#define DTRANK  64              // DMODEL/16
#define NTOK    (B_SZ * SEQ_L)  // 4096
#define DBC_W   (DTRANK + 2 * DSTATE)  // 96
#define LN_EPS  1e-5f

typedef float v2f __attribute__((ext_vector_type(2)));
typedef float v8f __attribute__((ext_vector_type(8)));

// -------------------------------------------------------------------------
// 1) LayerNorm: one token (1024 elems) per 256-thread block
// -------------------------------------------------------------------------
__global__ __launch_bounds__(256)
void ln_kernel(const float* __restrict__ x, const float* __restrict__ w,
               const float* __restrict__ b, float* __restrict__ xn) {
    __shared__ float s_sum[256];
    __shared__ float s_sq[256];
    const int tok = blockIdx.x;
    const float* xr = x + (size_t)tok * DMODEL;
    float v[4];
    float sum = 0.0f, sq = 0.0f;
#pragma unroll
    for (int i = 0; i < 4; ++i) {
        v[i] = xr[threadIdx.x + i * 256];
        sum += v[i];
        sq  += v[i] * v[i];
    }
    s_sum[threadIdx.x] = sum;
    s_sq[threadIdx.x]  = sq;
    __syncthreads();
    for (int off = 128; off > 0; off >>= 1) {
        if (threadIdx.x < off) {
            s_sum[threadIdx.x] += s_sum[threadIdx.x + off];
            s_sq[threadIdx.x]  += s_sq[threadIdx.x + off];
        }
        __syncthreads();
    }
    const float mu  = s_sum[0] * (1.0f / DMODEL);
    const float var = s_sq[0] * (1.0f / DMODEL) - mu * mu;
    const float inv = rsqrtf(var + LN_EPS);
    float* out = xn + (size_t)tok * DMODEL;
#pragma unroll
    for (int i = 0; i < 4; ++i) {
        const int c = threadIdx.x + i * 256;
        out[c] = (v[i] - mu) * inv * w[c] + b[c];
    }
}

// -------------------------------------------------------------------------
// 2a) LDS-staged, double-buffered fp32 WMMA GEMM for the large projections.
//     Block tile 64x128, 8 waves as 2x4, each wave owns a 32x32 register
//     tile (2x2 of 16x16 WMMA tiles).  2-chunk software pipeline:
//     chunk k+2's global b128 loads are issued between chunk k+1's LDS
//     stores and the barrier, so the compiler cannot sink them past the
//     fence — each load batch is covered by a barrier + 16 WMMAs.
//     MODE: 0 = none, 2 = +residual.
// -------------------------------------------------------------------------
#define BM 64
#define BN 128
#define BK 16
#define ALD (BK + 4)   // 20 words: rows 16B-aligned, 16 rows at stride 20 hit distinct banks mod 64

template <int MODE>
__global__ __launch_bounds__(256)
void gemm_wmma_f32_blk(const float* __restrict__ A, const float* __restrict__ B,
                       float* __restrict__ C, int M, int N, int K,
                       int lda, int ldb, int ldc,
                       const float* __restrict__ res, int ldr) {
    __shared__ float As[2][BM][ALD];
    __shared__ float Bs[2][BK][BN];

    const int lane = threadIdx.x & 31;
    const int wave = threadIdx.x >> 5;
    const int wrow = wave >> 2;        // 0..1
    const int wcol = wave & 3;         // 0..3
    const int half = lane >> 4;        // K-pair select per ISA A/B layout
    const int l16  = lane & 15;

    const int tilesN = N / BN;
    const int bm0 = (blockIdx.x / tilesN) * BM;
    const int bn0 = (blockIdx.x % tilesN) * BN;

    // global->LDS staging coordinates (b128 per load)
    const int ar = threadIdx.x >> 2;          // 0..63
    const int ak = (threadIdx.x & 3) * 4;     // 0,4,8,12
    const int br = threadIdx.x >> 5;          // 0..7 (row br and br+8)
    const int bc = (threadIdx.x & 31) * 4;    // 0..124

    const float* Aptr = &A[(size_t)(bm0 + ar) * lda + ak];
    const float* Bptr = &B[(size_t)br * ldb + bn0 + bc];

    const int arow0 = wrow * 32 + l16;
    const int arow1 = arow0 + 16;
    const int bcol0 = wcol * 32 + l16;
    const int bcol1 = bcol0 + 16;

    v8f acc[2][2] = {};

    // compute one BK chunk out of LDS buffer `p`
    auto compute = [&](int p) {
#pragma unroll
        for (int kk = 0; kk < BK; kk += 4) {
            const int kb = kk + 2 * half;
            v2f af[2], bf[2];
            af[0].x = As[p][arow0][kb];  af[0].y = As[p][arow0][kb + 1];
            af[1].x = As[p][arow1][kb];  af[1].y = As[p][arow1][kb + 1];
            bf[0].x = Bs[p][kb][bcol0];  bf[0].y = Bs[p][kb + 1][bcol0];
            bf[1].x = Bs[p][kb][bcol1];  bf[1].y = Bs[p][kb + 1][bcol1];
#pragma unroll
            for (int i = 0; i < 2; ++i) {
#pragma unroll
                for (int j = 0; j < 2; ++j) {
                    acc[i][j] = __builtin_amdgcn_wmma_f32_16x16x4_f32(
                        false, af[i], false, bf[j], (short)0, acc[i][j],
                        false, false);
                }
            }
        }
    };

    float4 ra, rb0, rb1;

    // prologue: stage chunk 0 into buffer 0, then issue chunk 1's loads
    {
        const float4 pa  = *(const float4*)(Aptr);
        const float4 pb0 = *(const float4*)(Bptr);
        const float4 pb1 = *(const float4*)(Bptr + (size_t)8 * ldb);
        *(float4*)&As[0][ar][ak]     = pa;
        *(float4*)&Bs[0][br][bc]     = pb0;
        *(float4*)&Bs[0][br + 8][bc] = pb1;
    }
    if (K > BK) {   // uniform
        ra  = *(const float4*)(Aptr + BK);
        rb0 = *(const float4*)(Bptr + (size_t)BK * ldb);
        rb1 = *(const float4*)(Bptr + (size_t)(BK + 8) * ldb);
    }
    __syncthreads();

    int buf = 0;
    for (int k0 = BK; k0 < K; k0 += BK) {
        compute(buf);          // 16 WMMAs on chunk (k0-BK)

        buf ^= 1;
        *(float4*)&As[buf][ar][ak]     = ra;    // chunk k0 -> LDS
        *(float4*)&Bs[buf][br][bc]     = rb0;
        *(float4*)&Bs[buf][br + 8][bc] = rb1;

        const int kn = k0 + BK;                 // issue chunk k0+BK loads;
        if (kn < K) {                           // fence below pins them here
            ra  = *(const float4*)(Aptr + kn);
            rb0 = *(const float4*)(Bptr + (size_t)kn * ldb);
            rb1 = *(const float4*)(Bptr + (size_t)(kn + 8) * ldb);
            if (kn + BK < K) {
                __builtin_prefetch(Bptr + (size_t)(kn + BK) * ldb, 0, 0);
            }
        }
        __syncthreads();
    }
    compute(buf);              // epilogue chunk

    // D layout: VGPR r -> row (+r + 8*half), col (+l16)
#pragma unroll
    for (int i = 0; i < 2; ++i) {
#pragma unroll
        for (int j = 0; j < 2; ++j) {
#pragma unroll
            for (int r = 0; r < 8; ++r) {
                const int row = bm0 + wrow * 32 + i * 16 + r + 8 * half;
                const int col = bn0 + wcol * 32 + j * 16 + l16;
                float vv = acc[i][j][r];
                if (MODE == 2) vv += res[(size_t)row * ldr + col];
                C[(size_t)row * ldc + col] = vv;
            }
        }
    }
}

// -------------------------------------------------------------------------
// 2b) simple fp32 WMMA GEMM (one 16x16 tile per wave) for small shapes
//     (x_proj N=96, dt_proj K=64).  MODE: 0 = none, 1 = softplus(.+bias[n])
// -------------------------------------------------------------------------
template <int MODE>
__global__ __launch_bounds__(256)
void gemm_wmma_f32(const float* __restrict__ A, const float* __restrict__ B,
                   float* __restrict__ C, int M, int N, int K,
                   int lda, int ldb, int ldc,
                   const float* __restrict__ bias) {
    const int lane  = threadIdx.x & 31;
    const int wave  = threadIdx.x >> 5;
    const int tilesN = N >> 4;
    const int tile  = blockIdx.x * 8 + wave;
    const int tm    = (tile / tilesN) << 4;
    const int tn    = (tile % tilesN) << 4;
    const int half  = lane >> 4;
    const int l16   = lane & 15;

    const float* arow = A + (size_t)(tm + l16) * lda + 2 * half;
    const float* bcol = B + (size_t)(2 * half) * ldb + tn + l16;

    v8f acc = {};
    for (int k = 0; k < K; k += 4) {
        v2f a, b;
        a.x = arow[k];
        a.y = arow[k + 1];
        b.x = bcol[(size_t)k * ldb];
        b.y = bcol[(size_t)(k + 1) * ldb];
        acc = __builtin_amdgcn_wmma_f32_16x16x4_f32(
            false, a, false, b, (short)0, acc, false, false);
    }

#pragma unroll
    for (int r = 0; r < 8; ++r) {
        const int row = tm + r + 8 * half;
        const int col = tn + l16;
        float vv = acc[r];
        if (MODE == 1) {                  // softplus(v + bias[col])
            const float t = vv + bias[col];
            vv = (t > 20.0f) ? t : log1pf(__expf(t));
        }
        C[(size_t)row * ldc + col] = vv;
    }
}

// -------------------------------------------------------------------------
// 3) causal depthwise conv (width 4) + bias + SiLU over the xi half of xz
// -------------------------------------------------------------------------
__global__ __launch_bounds__(256)
void conv_silu_kernel(const float* __restrict__ xz, const float* __restrict__ cw,
                      const float* __restrict__ cb, float* __restrict__ xi) {
    const int idx = blockIdx.x * blockDim.x + threadIdx.x;  // NTOK*DINNER
    const int d   = idx & (DINNER - 1);
    const int tok = idx >> 11;           // DINNER == 2048
    const int b   = tok >> 11;           // SEQ_L == 2048
    const int t   = tok & (SEQ_L - 1);
    float acc = cb[d];
#pragma unroll
    for (int j = 0; j < DCONV; ++j) {
        const int tt = t - (DCONV - 1) + j;
        if (tt >= 0) {
            acc += xz[((size_t)(b * SEQ_L + tt) * (2 * DINNER)) + d] *
                   cw[d * DCONV + j];
        }
    }
    xi[idx] = acc / (1.0f + __expf(-acc));  // SiLU
}

// -------------------------------------------------------------------------
// 4) selective scan: wave32 = 2 channels x 16 states; in-register
//    cross-state reduce via shfl_xor; fused +u*D and *silu(z) gate.
// -------------------------------------------------------------------------
__global__ __launch_bounds__(256)
void scan_kernel(const float* __restrict__ delta, const float* __restrict__ xi,
                 const float* __restrict__ dbc,   const float* __restrict__ xz,
                 const float* __restrict__ A_log, const float* __restrict__ Dp,
                 float* __restrict__ yg) {
    const int lane = threadIdx.x & 31;
    const int wgl  = (blockIdx.x * blockDim.x + threadIdx.x) >> 5;  // global wave
    const int n    = lane & 15;
    const int c    = wgl * 2 + (lane >> 4);   // channel 0..B_SZ*DINNER-1
    const int b    = c >> 11;                 // DINNER == 2048
    const int d    = c & (DINNER - 1);

    const float An = -__expf(A_log[d * DSTATE + n]);
    const float Dd = Dp[d];
    float h = 0.0f;
    const size_t tok0 = (size_t)b * SEQ_L;

    for (int t = 0; t < SEQ_L; ++t) {
        const size_t tok = tok0 + t;
        const float dv = delta[tok * DINNER + d];
        const float u  = xi[tok * DINNER + d];
        const float Bt = dbc[tok * DBC_W + DTRANK + n];
        const float Ct = dbc[tok * DBC_W + DTRANK + DSTATE + n];
        h = h * __expf(dv * An) + (dv * u) * Bt;
        float p = h * Ct;
        p += __shfl_xor(p, 1, 32);
        p += __shfl_xor(p, 2, 32);
        p += __shfl_xor(p, 4, 32);
        p += __shfl_xor(p, 8, 32);   // each 16-lane group now holds the sum
        if (n == 0) {
            const float y = p + u * Dd;
            const float z = xz[tok * (2 * DINNER) + DINNER + d];
            const float g = z / (1.0f + __expf(-z));
            yg[tok * DINNER + d] = y * g;
        }
    }
}

// -------------------------------------------------------------------------
// launcher
// -------------------------------------------------------------------------
extern "C" void kernel_launch(void* const* d_in, const int* in_sizes, int n_in,
                              void* d_out, int out_size, void* d_ws, size_t ws_size,
                              hipStream_t stream) {
    const float* x          = (const float*)d_in[0];
    const float* ln_w       = (const float*)d_in[1];
    const float* ln_b       = (const float*)d_in[2];
    const float* in_proj_w  = (const float*)d_in[3];   // [1024, 4096]
    const float* conv_w     = (const float*)d_in[4];   // [2048, 4]
    const float* conv_b     = (const float*)d_in[5];   // [2048]
    const float* x_proj_w   = (const float*)d_in[6];   // [2048, 96]
    const float* dt_proj_w  = (const float*)d_in[7];   // [64, 2048]
    const float* dt_proj_b  = (const float*)d_in[8];   // [2048]
    const float* A_log      = (const float*)d_in[9];   // [2048, 16]
    const float* D_param    = (const float*)d_in[10];  // [2048]
    const float* out_proj_w = (const float*)d_in[11];  // [2048, 1024]
    float* out = (float*)d_out;

    // workspace carve-up (fp32), total ~186 MB — L2-resident on MI455X
    float* ws     = (float*)d_ws;
    float* xn     = ws;                                  // 4096*1024
    float* xz     = xn    + (size_t)NTOK * DMODEL;       // 4096*4096
    float* xi_act = xz    + (size_t)NTOK * 2 * DINNER;   // 4096*2048
    float* dbc    = xi_act + (size_t)NTOK * DINNER;      // 4096*96
    float* delta  = dbc   + (size_t)NTOK * DBC_W;        // 4096*2048
    float* yg     = delta + (size_t)NTOK * DINNER;       // 4096*2048

    // 1) layernorm
    ln_kernel<<<NTOK, 256, 0, stream>>>(x, ln_w, ln_b, xn);

    // 2) in_proj: xz[4096,4096] = xn[4096,1024] @ in_proj_w[1024,4096]
    {
        const int M = NTOK, N = 2 * DINNER, K = DMODEL;
        const int blocks = (M / BM) * (N / BN);          // 64*32 = 2048
        gemm_wmma_f32_blk<0><<<blocks, 256, 0, stream>>>(
            xn, in_proj_w, xz, M, N, K, K, N, N, nullptr, 0);
    }

    // 3) depthwise causal conv + SiLU on the xi half of xz
    conv_silu_kernel<<<(NTOK * DINNER) / 256, 256, 0, stream>>>(
        xz, conv_w, conv_b, xi_act);

    // 4) x_proj: dbc[4096,96] = xi_act[4096,2048] @ x_proj_w[2048,96]
    {
        const int M = NTOK, N = DBC_W, K = DINNER;
        const int blocks = (M / 16) * (N / 16) / 8;      // 192
        gemm_wmma_f32<0><<<blocks, 256, 0, stream>>>(
            xi_act, x_proj_w, dbc, M, N, K, K, N, N, nullptr);
    }

    // 5) dt_proj + softplus: delta[4096,2048] =
    //    softplus(dbc[:, :64] @ dt_proj_w[64,2048] + dt_proj_b)
    {
        const int M = NTOK, N = DINNER, K = DTRANK;
        const int blocks = (M / 16) * (N / 16) / 8;      // 4096
        gemm_wmma_f32<1><<<blocks, 256, 0, stream>>>(
            dbc, dt_proj_w, delta, M, N, K, DBC_W, N, N, dt_proj_b);
    }

    // 6) selective scan + D skip + SiLU(z) gate -> yg[4096,2048]
    scan_kernel<<<(B_SZ * DINNER * 16) / 256, 256, 0, stream>>>(
        delta, xi_act, dbc, xz, A_log, D_param, yg);

    // 7) out_proj + residual: out[4096,1024] = yg @ out_proj_w + x
    {
        const int M = NTOK, N = DMODEL, K = DINNER;
        const int blocks = (M / BM) * (N / BN);          // 64*8 = 512
        gemm_wmma_f32_blk<2><<<blocks, 256, 0, stream>>>(
            yg, out_proj_w, out, M, N, K, K, N, N, x, DMODEL);
    }
}